// GPUHierarchy_33913061769276
// MI455X (gfx1250) — compile-verified
//
#include <hip/hip_runtime.h>

#define MAX_AGENTS   64
#define IN_DIM       256
#define HID_DIM      512
#define OUT_DIM      256
#define MAX_CHILDREN 8
#define BATCH        1024

typedef float v2f __attribute__((ext_vector_type(2)));
typedef float v8f __attribute__((ext_vector_type(8)));

// ---------------- scheduling kernels (group samples by agent) ----------------

__global__ void moe_zero_counts(int* __restrict__ cnt) {
    int t = threadIdx.x;
    if (t < MAX_AGENTS) cnt[t] = 0;
}

__global__ void moe_count(const int* __restrict__ agents, int* __restrict__ cnt) {
    int i = blockIdx.x * blockDim.x + threadIdx.x;
    if (i < BATCH) atomicAdd(&cnt[agents[i]], 1);
}

__global__ void moe_scan(const int* __restrict__ cnt,
                         int* __restrict__ off, int* __restrict__ cur) {
    if (threadIdx.x == 0) {
        int s = 0;
        for (int a = 0; a < MAX_AGENTS; ++a) { off[a] = s; cur[a] = s; s += cnt[a]; }
    }
}

__global__ void moe_scatter(const int* __restrict__ agents,
                            int* __restrict__ cur, int* __restrict__ perm) {
    int i = blockIdx.x * blockDim.x + threadIdx.x;
    if (i < BATCH) {
        int p = atomicAdd(&cur[agents[i]], 1);
        perm[p] = i;
    }
}

// ---------------- main fused MoE kernel: one workgroup per agent ----------------
// 256 threads = 8 wave32s. LDS: X tile (16x256, padded) + H tile (16x512, padded).
// Row stride padded by +4 floats so A-fragment column reads hit distinct banks.

__global__ __launch_bounds__(256)
void moe_agent_mlp(const float* __restrict__ x,
                   const float* __restrict__ W1, const float* __restrict__ b1,
                   const float* __restrict__ W2, const float* __restrict__ b2,
                   const float* __restrict__ rw,
                   const int* __restrict__ cnt, const int* __restrict__ off,
                   const int* __restrict__ perm,
                   float* __restrict__ h_out, float* __restrict__ o_out,
                   float* __restrict__ r_out)
{
    __shared__ float Xs[16][IN_DIM + 4];    // X tile; reused as Out staging (cols < 256)
    __shared__ float Hs[16][HID_DIM + 4];   // H tile
    __shared__ int   rows[16];

    const int a = blockIdx.x;
    const int n = cnt[a];
    if (n == 0) return;                      // uniform per block
    const int base = off[a];

    const int t     = threadIdx.x;
    const int lane  = t & 31;
    const int wave  = t >> 5;
    const int colN  = lane & 15;             // N column (B/C/D) and M row (A) for this lane
    const int kb    = (lane >> 4) << 1;      // K sub-pair base: 0 or 2
    const int rhalf = (lane >> 4) << 3;      // C/D row half: 0 or 8

    const float* W1a = W1 + (size_t)a * IN_DIM * HID_DIM;
    const float* W2a = W2 + (size_t)a * HID_DIM * OUT_DIM;
    const float* b1a = b1 + (size_t)a * HID_DIM;
    const float* b2a = b2 + (size_t)a * OUT_DIM;
    const float* rwa = rw + (size_t)a * HID_DIM * MAX_CHILDREN;

    const int mtiles = (n + 15) >> 4;
    for (int mt = 0; mt < mtiles; ++mt) {
        // 1) sample row ids for this 16-row tile (pad replays first sample: identical result)
        if (t < 16) {
            int s = base + mt * 16 + t;
            rows[t] = (s < base + n) ? perm[s] : perm[base];
        }
        __syncthreads();

        // 2) stage X tile into LDS (coalesced 16x256)
        for (int idx = t; idx < 16 * IN_DIM; idx += 256) {
            int r = idx >> 8;
            int c = idx & (IN_DIM - 1);
            Xs[r][c] = x[(size_t)rows[r] * IN_DIM + c];
        }
        __syncthreads();

        // 3) GEMM1: H = relu(X * W1 + b1). 32 N-tiles of 16 cols, 4 per wave.
        for (int i = 0; i < 4; ++i) {
            const int   gcol = (wave * 4 + i) * 16 + colN;
            const float bias = b1a[gcol];
            v8f acc = { bias, bias, bias, bias, bias, bias, bias, bias };
            const float* bp0 = W1a + (size_t)kb * HID_DIM + gcol;
            #pragma unroll 8
            for (int k = 0; k < IN_DIM / 4; ++k) {
                v2f av;
                av.x = Xs[colN][k * 4 + kb];
                av.y = Xs[colN][k * 4 + kb + 1];
                const float* bp = bp0 + (size_t)(k * 4) * HID_DIM;
                v2f bv;
                bv.x = bp[0];
                bv.y = bp[HID_DIM];
                acc = __builtin_amdgcn_wmma_f32_16x16x4_f32(
                          false, av, false, bv, (short)0, acc, false, false);
            }
            #pragma unroll
            for (int v = 0; v < 8; ++v) {
                float hv = acc[v];
                Hs[rhalf + v][gcol] = hv > 0.0f ? hv : 0.0f;   // fused ReLU
            }
        }
        __syncthreads();

        // 4) write H tile to global (coalesced)
        for (int idx = t; idx < 16 * HID_DIM; idx += 256) {
            int r = idx >> 9;
            int c = idx & (HID_DIM - 1);
            h_out[(size_t)rows[r] * HID_DIM + c] = Hs[r][c];
        }

        // 5) GEMM2: Out = H * W2 + b2. 16 N-tiles, 2 per wave. Stage into Xs.
        for (int i = 0; i < 2; ++i) {
            const int   gcol = (wave * 2 + i) * 16 + colN;
            const float bias = b2a[gcol];
            v8f acc = { bias, bias, bias, bias, bias, bias, bias, bias };
            const float* bp0 = W2a + (size_t)kb * OUT_DIM + gcol;
            #pragma unroll 8
            for (int k = 0; k < HID_DIM / 4; ++k) {
                v2f av;
                av.x = Hs[colN][k * 4 + kb];
                av.y = Hs[colN][k * 4 + kb + 1];
                const float* bp = bp0 + (size_t)(k * 4) * OUT_DIM;
                v2f bv;
                bv.x = bp[0];
                bv.y = bp[OUT_DIM];
                acc = __builtin_amdgcn_wmma_f32_16x16x4_f32(
                          false, av, false, bv, (short)0, acc, false, false);
            }
            #pragma unroll
            for (int v = 0; v < 8; ++v) {
                Xs[rhalf + v][gcol] = acc[v];
            }
        }
        __syncthreads();

        // 6) write Out tile to global (coalesced)
        for (int idx = t; idx < 16 * OUT_DIM; idx += 256) {
            int r = idx >> 8;
            int c = idx & (OUT_DIM - 1);
            o_out[(size_t)rows[r] * OUT_DIM + c] = Xs[r][c];
        }

        // 7) router logits: 16 rows x 8 children via VALU dot products over H in LDS
        if (t < 128) {
            int rr = t >> 3, c = t & 7;
            float s = 0.0f;
            for (int kk = 0; kk < HID_DIM; ++kk)
                s = fmaf(Hs[rr][kk], rwa[(size_t)kk * MAX_CHILDREN + c], s);
            r_out[(size_t)rows[rr] * MAX_CHILDREN + c] = s;
        }
        __syncthreads();
    }
}

// ---------------- launch ----------------

extern "C" void kernel_launch(void* const* d_in, const int* in_sizes, int n_in,
                              void* d_out, int out_size, void* d_ws, size_t ws_size,
                              hipStream_t stream) {
    const float* x      = (const float*)d_in[0];
    const float* W1     = (const float*)d_in[1];
    const float* b1     = (const float*)d_in[2];
    const float* W2     = (const float*)d_in[3];
    const float* b2     = (const float*)d_in[4];
    const float* rw     = (const float*)d_in[5];
    const int*   agents = (const int*)d_in[6];

    float* out   = (float*)d_out;
    float* h_out = out;                                   // [1024, 512]
    float* o_out = out + (size_t)BATCH * HID_DIM;         // [1024, 256]
    float* r_out = o_out + (size_t)BATCH * OUT_DIM;       // [1024, 8]

    int* cnt  = (int*)d_ws;
    int* off  = cnt + MAX_AGENTS;
    int* cur  = off + MAX_AGENTS;
    int* perm = cur + MAX_AGENTS;                         // [BATCH]

    moe_zero_counts<<<1, 64, 0, stream>>>(cnt);
    moe_count<<<BATCH / 256, 256, 0, stream>>>(agents, cnt);
    moe_scan<<<1, 1, 0, stream>>>(cnt, off, cur);
    moe_scatter<<<BATCH / 256, 256, 0, stream>>>(agents, cur, perm);
    moe_agent_mlp<<<MAX_AGENTS, 256, 0, stream>>>(x, W1, b1, W2, b2, rw,
                                                  cnt, off, perm,
                                                  h_out, o_out, r_out);
}